// DualLossDiscrete_76416058130899
// MI455X (gfx1250) — compile-verified
//
#include <hip/hip_runtime.h>

// ---------------------------------------------------------------------------
// DualLossDiscrete on MI455X (gfx1250, wave32)
//
//   diff = eq_transform(edge_inv_global - d_target, pos_perturbed, ...)
//   out  = 10 * sum(diff^2) / (3N)
//
// Pass 1 (gd_init):   zero N*3 accumulator + f64 total in workspace
// Pass 2 (gd_edges):  1 thread/edge; streaming edge data read non-temporal
//                     (TH=NT) so the ~16MB gather set (pos/posp/n2g/sc/acc)
//                     stays resident in 192MB L2; 6x global_atomic_add_f32
// Pass 3 (gd_reduce): b128 NT loads, sum-of-squares; wave reduction via
//                     v_wmma_f32_16x16x4_f32 (B = ones -> row sums),
//                     f64 across waves/blocks
// Pass 4 (gd_final):  scale + write scalar
// ---------------------------------------------------------------------------

typedef __attribute__((ext_vector_type(2))) float v2f;
typedef __attribute__((ext_vector_type(4))) float v4f;
typedef __attribute__((ext_vector_type(8))) float v8f;

struct F3 { float x, y, z; };

__global__ void gd_init(float* __restrict__ acc, double* __restrict__ total, int n3) {
  int i = blockIdx.x * blockDim.x + threadIdx.x;
  if (i < n3) acc[i] = 0.0f;
  if (i == 0) *total = 0.0;
}

__global__ void gd_edges(const float* __restrict__ edge_inv,
                         const float* __restrict__ edge_len,
                         const float* __restrict__ alpha,
                         const F3* __restrict__ pos,
                         const F3* __restrict__ posp,
                         const int* __restrict__ ei,
                         const int* __restrict__ n2g,
                         const unsigned char* __restrict__ sc,
                         float* __restrict__ acc,
                         int E) {
  int e = blockIdx.x * blockDim.x + threadIdx.x;
  if (e >= E) return;

  // Single-use streaming reads: non-temporal so they don't evict the hot
  // gather set from L2.
  const int   row = __builtin_nontemporal_load(ei + e);
  const int   col = __builtin_nontemporal_load(ei + E + e);
  const float el  = __builtin_nontemporal_load(edge_len + e);
  const float inv = __builtin_nontemporal_load(edge_inv + e);

  // ground-truth distance from pos (hot gather, regular temporal)
  F3 pr = pos[row];
  F3 pc = pos[col];
  float dx = pr.x - pc.x, dy = pr.y - pc.y, dz = pr.z - pc.z;
  float dgt = sqrtf(dx * dx + dy * dy + dz * dz);

  // d_target = (d_gt - d_perturbed) * sqrt(a/(1-a)); zero when mask is false
  bool mask = (sc[row] | sc[col]) != 0;
  float ae  = alpha[n2g[row]];
  float dtg = mask ? (dgt - el) * sqrtf(ae / (1.0f - ae)) : 0.0f;

  // combined edge score (linearity of eq_transform)
  float s = (inv - dtg) / el;

  F3 qr = posp[row];
  F3 qc = posp[col];
  float mx = s * (qr.x - qc.x);
  float my = s * (qr.y - qc.y);
  float mz = s * (qr.z - qc.z);

  float* ar = acc + 3 * row;
  float* ac = acc + 3 * col;
  unsafeAtomicAdd(ar + 0,  mx);
  unsafeAtomicAdd(ar + 1,  my);
  unsafeAtomicAdd(ar + 2,  mz);
  unsafeAtomicAdd(ac + 0, -mx);
  unsafeAtomicAdd(ac + 1, -my);
  unsafeAtomicAdd(ac + 2, -mz);
}

__global__ void gd_reduce(const float* __restrict__ acc,
                          double* __restrict__ total, int n3) {
  float p = 0.0f;
  const int tid    = blockIdx.x * blockDim.x + threadIdx.x;
  const int stride = gridDim.x * blockDim.x;

  // Vectorized body: b128 non-temporal loads (acc is dead after this pass).
  const int n4 = n3 >> 2;
  const v4f* acc4 = (const v4f*)acc;
  for (int i = tid; i < n4; i += stride) {
    v4f v = __builtin_nontemporal_load(acc4 + i);
    p = fmaf(v[0], v[0], p);
    p = fmaf(v[1], v[1], p);
    p = fmaf(v[2], v[2], p);
    p = fmaf(v[3], v[3], p);
  }
  // Scalar tail.
  for (int i = (n4 << 2) + tid; i < n3; i += stride) {
    float v = __builtin_nontemporal_load(acc + i);
    p = fmaf(v, v, p);
  }

  // Wave32 reduction via WMMA: A(16x4) holds per-lane partials
  //   lanes 0-15:  A[m][0] = p_m        (VGPR0), A[m][1] = 0 (VGPR1)
  //   lanes 16-31: A[m][2] = p_{m+16},  A[m][3] = 0
  // B(4x16) = ones  =>  D[m][n] = p_m + p_{m+16}
  v2f A; A[0] = p;    A[1] = 0.0f;
  v2f B; B[0] = 1.0f; B[1] = 1.0f;
  v8f C = {};
  v8f D = __builtin_amdgcn_wmma_f32_16x16x4_f32(
      /*neg_a=*/false, A, /*neg_b=*/false, B,
      /*c_mod=*/(short)0, C, /*reuse_a=*/false, /*reuse_b=*/false);

  // lane L<16  : sum_j D[j] = sum_{m=0..7}  (p_m + p_{m+16})
  // lane L>=16 : sum_j D[j] = sum_{m=8..15} (p_m + p_{m+16})
  float t = 0.0f;
#pragma unroll
  for (int j = 0; j < 8; ++j) t += D[j];
  t += __shfl_down(t, 16, 32);   // lane 0 now has the wave total

  __shared__ double partial[32];
  int lane = threadIdx.x & 31;
  int wave = threadIdx.x >> 5;
  if (lane == 0) partial[wave] = (double)t;
  __syncthreads();
  if (threadIdx.x == 0) {
    double sblk = 0.0;
    int nw = (blockDim.x + 31) >> 5;
    for (int w = 0; w < nw; ++w) sblk += partial[w];
    unsafeAtomicAdd(total, sblk);
  }
}

__global__ void gd_final(const double* __restrict__ total,
                         float* __restrict__ out, double scale) {
  out[0] = (float)(*total * scale);
}

extern "C" void kernel_launch(void* const* d_in, const int* in_sizes, int n_in,
                              void* d_out, int out_size, void* d_ws, size_t ws_size,
                              hipStream_t stream) {
  const float* edge_inv     = (const float*)d_in[0];         // [E,1]
  const float* edge_len     = (const float*)d_in[1];         // [E,1]
  const float* alpha        = (const float*)d_in[2];         // [G]
  const F3*    pos          = (const F3*)d_in[3];            // [N,3]
  const F3*    posp         = (const F3*)d_in[4];            // [N,3]
  const int*   ei           = (const int*)d_in[5];           // [2,E]
  const int*   n2g          = (const int*)d_in[6];           // [N]
  const unsigned char* sc   = (const unsigned char*)d_in[7]; // [N] bool
  float* out = (float*)d_out;

  const int E  = in_sizes[0];
  const int N  = in_sizes[3] / 3;
  const int n3 = 3 * N;

  float* acc = (float*)d_ws;
  size_t off = (((size_t)n3 * sizeof(float)) + 15) & ~(size_t)15;
  double* total = (double*)((char*)d_ws + off);

  const int B = 256;
  gd_init <<<(n3 + B - 1) / B, B, 0, stream>>>(acc, total, n3);
  gd_edges<<<(E  + B - 1) / B, B, 0, stream>>>(edge_inv, edge_len, alpha, pos,
                                               posp, ei, n2g, sc, acc, E);
  gd_reduce<<<1024, B, 0, stream>>>(acc, total, n3);
  gd_final <<<1, 1, 0, stream>>>(total, out, 10.0 / (double)n3);
}